// SectorWiseAgg_60026462929394
// MI455X (gfx1250) — compile-verified
//
#include <hip/hip_runtime.h>
#include <math.h>

// ---- problem constants (match reference) ----
#define NN   20000   // nodes
#define IND  256     // in_dim
#define OUTD 128     // out_dim
#define SS   8       // sectors
#define KK   9       // branches (self + sectors)
#define EP   40000   // edges per sector

typedef __attribute__((ext_vector_type(2))) float v2f;
typedef __attribute__((ext_vector_type(8))) float v8f;

static __device__ __forceinline__ v8f wmma_f32(v2f a, v2f b, v8f c) {
    // D = A(16x4) * B(4x16) + C(16x16), fp32 exact
    return __builtin_amdgcn_wmma_f32_16x16x4_f32(false, a, false, b, (short)0, c,
                                                 false, false);
}

// ---------------------------------------------------------------------------
// zero helper
// ---------------------------------------------------------------------------
__global__ void zero_f4_kernel(float4* __restrict__ p, long n4) {
    long i = (long)blockIdx.x * blockDim.x + threadIdx.x;
    if (i < n4) p[i] = make_float4(0.f, 0.f, 0.f, 0.f);
}

// ---------------------------------------------------------------------------
// Batched GEMM: C[b] = rowScale * (A @ W[b]),   A: [M x Kdim], W: [Kdim x 128]
// 8 waves / block, each wave owns a 16-row strip and all 8 column tiles.
// W panel staged through LDS in 64x128 chunks (32 KB), stored k-pair
// interleaved: wlds[p*256 + col*2 + r] = W[kc + 2p + r][col], so each WMMA
// B-fragment is a single aligned ds_load_b64 (no repacking movs).
// ---------------------------------------------------------------------------
__global__ __launch_bounds__(256)
void gemm_wmma_kernel(const float* __restrict__ A, int lda,
                      const float* __restrict__ Wbase, long wBatchStride,
                      float* __restrict__ Cbase, long cBatchStride, int ldc,
                      const float* __restrict__ rowScale, int scaleDiv,
                      int M, int Kdim) {
    __shared__ float wlds[64 * 128];   // 32 KB, pair-interleaved

    const int tid  = threadIdx.x;
    const int wave = tid >> 5;
    const int lane = tid & 31;
    const int cl   = lane & 15;
    const int half = lane >> 4;

    const float* W = Wbase + (long)blockIdx.y * wBatchStride;
    float*       C = Cbase + (long)blockIdx.y * cBatchStride;

    const int strip = blockIdx.x * 8 + wave;
    const int r0    = strip * 16;
    const bool active = (r0 < M);          // uniform per wave -> EXEC all-ones inside

    v8f acc[8] = {};

    for (int kc = 0; kc < Kdim; kc += 64) {
        __syncthreads();
        // cooperative stage: interleave k-pairs while copying (same traffic as
        // a straight copy: 2x global_load_b128 + 2x 16B LDS store per task)
        #pragma unroll
        for (int it = tid; it < 1024; it += 256) {
            const int p  = it >> 5;          // k-pair index 0..31
            const int c0 = (it & 31) * 4;    // column 0..124 step 4
            const float4 ra = *(const float4*)(W + (size_t)(kc + 2 * p) * 128 + c0);
            const float4 rb = *(const float4*)(W + (size_t)(kc + 2 * p + 1) * 128 + c0);
            float4* dst = (float4*)&wlds[p * 256 + c0 * 2];
            dst[0] = make_float4(ra.x, rb.x, ra.y, rb.y);
            dst[1] = make_float4(ra.z, rb.z, ra.w, rb.w);
        }
        __syncthreads();

        if (active) {
            const float* arow = A + (size_t)(r0 + cl) * lda + kc;
            #pragma unroll
            for (int ks = 0; ks < 16; ++ks) {
                const int kl = ks * 4 + 2 * half;    // even k within chunk
                const int p  = kl >> 1;              // pair index
                v2f a = *(const v2f*)(arow + kl);    // A[m, kk], A[m, kk+1]
                const float* brow = &wlds[p * 256 + cl * 2];
                #pragma unroll
                for (int t = 0; t < 8; ++t) {
                    v2f b = *(const v2f*)(brow + t * 32);   // {W[kk][col], W[kk+1][col]}
                    acc[t] = wmma_f32(a, b, acc[t]);
                }
            }
        }
    }

    if (active) {
        float s8[8];
        #pragma unroll
        for (int i = 0; i < 8; ++i) {
            int row = r0 + i + half * 8;
            s8[i] = rowScale ? rowScale[row / scaleDiv] : 1.0f;
        }
        #pragma unroll
        for (int t = 0; t < 8; ++t) {
            #pragma unroll
            for (int i = 0; i < 8; ++i) {
                int row = r0 + i + half * 8;
                C[(size_t)row * ldc + t * 16 + cl] = acc[t][i] * s8[i];
            }
        }
    }
}

// ---------------------------------------------------------------------------
// Edge aggregation: z[src, s+1, :] += H[s, dst, :]   (one wave per edge)
// ---------------------------------------------------------------------------
__global__ __launch_bounds__(256)
void agg_kernel(const float* __restrict__ H, const int* __restrict__ rows,
                const int* __restrict__ cols, float* __restrict__ z) {
    const int wave = threadIdx.x >> 5;
    const int lane = threadIdx.x & 31;
    const long idx = (long)blockIdx.x * 8 + wave;     // edge id over S*EP
    if (idx >= (long)SS * EP) return;                 // wave-uniform
    const int s = (int)(idx / EP);
    const int r = rows[idx];
    const int c = cols[idx];
    const float* src = H + ((size_t)s * NN + c) * OUTD;
    float*       dst = z + ((size_t)r * KK + s + 1) * OUTD;
    #pragma unroll
    for (int i = 0; i < 4; ++i)
        atomicAdd(&dst[lane + 32 * i], src[lane + 32 * i]);
}

// ---------------------------------------------------------------------------
// Per-node interaction: grams + softmax + attn-apply + gate, one wave / node.
// zc/zd rows padded 9 -> 16 with zeros so WMMA padding contributes nothing.
// ---------------------------------------------------------------------------
__global__ __launch_bounds__(32)
void interact_kernel(const float* __restrict__ zc_g, const float* __restrict__ zd_g,
                     const float* __restrict__ gate_w, const float* __restrict__ gate_b,
                     float* __restrict__ out) {
    __shared__ float zc_s[16 * 128];
    __shared__ float zd_s[16 * 128];
    __shared__ float ga[16 * 16];       // gram of zc -> softmax in place -> ac
    __shared__ float gd[16 * 16];       // gram of zd -> softmax in place -> ad
    __shared__ float zcom[16 * 128];
    __shared__ float zdis[16 * 128];

    const int n   = blockIdx.x;
    const int tid = threadIdx.x;
    const int cl  = tid & 15;
    const int half = tid >> 4;

    // load zc/zd (rows 0..8 real, 9..15 zero) as float4
    const float4* zc4 = (const float4*)(zc_g + (size_t)n * KK * 128);
    const float4* zd4 = (const float4*)(zd_g + (size_t)n * KK * 128);
    #pragma unroll
    for (int i = tid; i < 16 * 32; i += 32) {
        const int row = i >> 5;
        float4 vz = make_float4(0.f, 0.f, 0.f, 0.f);
        float4 wz = make_float4(0.f, 0.f, 0.f, 0.f);
        if (row < KK) { vz = zc4[i]; wz = zd4[i]; }
        ((float4*)zc_s)[i] = vz;
        ((float4*)zd_s)[i] = wz;
    }
    __syncthreads();

    // grams: G = Z @ Z^T  (A-frag == B-frag by symmetry of the layouts)
    v8f gacc = {}, dacc = {};
    #pragma unroll 8
    for (int k0 = 0; k0 < 128; k0 += 4) {
        const int kk = k0 + 2 * half;
        v2f a = *(const v2f*)(&zc_s[cl * 128 + kk]);
        gacc = wmma_f32(a, a, gacc);
        v2f b = *(const v2f*)(&zd_s[cl * 128 + kk]);
        dacc = wmma_f32(b, b, dacc);
    }
    #pragma unroll
    for (int i = 0; i < 8; ++i) {
        const int row = i + half * 8;
        ga[row * 16 + cl] = gacc[i];
        gd[row * 16 + cl] = dacc[i];
    }
    __syncthreads();

    // row softmaxes (9 valid entries), in place; padding stays exactly 0
    if (tid < KK) {
        float v[KK];
        float m = -3.402823466e38f;
        #pragma unroll
        for (int j = 0; j < KK; ++j) { v[j] = ga[tid * 16 + j]; m = fmaxf(m, v[j]); }
        float s = 0.f;
        #pragma unroll
        for (int j = 0; j < KK; ++j) { v[j] = __expf(v[j] - m); s += v[j]; }
        float inv = 1.0f / s;
        #pragma unroll
        for (int j = 0; j < KK; ++j) ga[tid * 16 + j] = v[j] * inv;

        const float dkk = gd[tid * 16 + tid];   // ds[k,j] = G[k,k] - G[k,j]
        m = -3.402823466e38f;
        #pragma unroll
        for (int j = 0; j < KK; ++j) { v[j] = dkk - gd[tid * 16 + j]; m = fmaxf(m, v[j]); }
        s = 0.f;
        #pragma unroll
        for (int j = 0; j < KK; ++j) { v[j] = __expf(v[j] - m); s += v[j]; }
        inv = 1.0f / s;
        #pragma unroll
        for (int j = 0; j < KK; ++j) gd[tid * 16 + j] = v[j] * inv;
    }
    __syncthreads();

    // z_com = ac @ zc ; z_dis = zd - ad @ zd   (softmax rows sum to 1)
    #pragma unroll
    for (int t = 0; t < 8; ++t) {
        v8f ca = {}, da = {};
        #pragma unroll
        for (int k0 = 0; k0 < 16; k0 += 4) {
            const int kk = k0 + 2 * half;
            v2f a  = *(const v2f*)(&ga[cl * 16 + kk]);
            v2f a2 = *(const v2f*)(&gd[cl * 16 + kk]);
            v2f b, b2;
            b.x  = zc_s[kk * 128 + t * 16 + cl];
            b.y  = zc_s[(kk + 1) * 128 + t * 16 + cl];
            b2.x = zd_s[kk * 128 + t * 16 + cl];
            b2.y = zd_s[(kk + 1) * 128 + t * 16 + cl];
            ca = wmma_f32(a, b, ca);
            da = wmma_f32(a2, b2, da);
        }
        #pragma unroll
        for (int i = 0; i < 8; ++i) {
            const int row = i + half * 8;
            const int col = t * 16 + cl;
            zcom[row * 128 + col] = ca[i];
            zdis[row * 128 + col] = zd_s[row * 128 + col] - da[i];
        }
    }
    __syncthreads();

    // gate: beta = sigmoid([z_com, z_dis] . gate_w + b)
    float part = 0.f;
    for (int idx = tid; idx < KK * 128; idx += 32)
        part += zcom[idx] * gate_w[idx] + zdis[idx] * gate_w[KK * 128 + idx];
    #pragma unroll
    for (int off = 16; off > 0; off >>= 1) part += __shfl_xor(part, off, 32);
    const float beta = 1.0f / (1.0f + __expf(-(part + gate_b[0])));

    float* o = out + (size_t)n * (KK * 128);
    for (int idx = tid; idx < KK * 128; idx += 32)
        o[idx] = beta * zcom[idx] + (1.0f - beta) * zdis[idx];
}

// ---------------------------------------------------------------------------
extern "C" void kernel_launch(void* const* d_in, const int* in_sizes, int n_in,
                              void* d_out, int out_size, void* d_ws, size_t ws_size,
                              hipStream_t stream) {
    const float* x        = (const float*)d_in[0];
    const float* W_self   = (const float*)d_in[1];
    const float* W_sect   = (const float*)d_in[2];
    const float* WC       = (const float*)d_in[3];
    const float* WD       = (const float*)d_in[4];
    const float* gate_w   = (const float*)d_in[5];
    const float* gate_b   = (const float*)d_in[6];
    const float* out_norm = (const float*)d_in[7];
    const float* in_norm  = (const float*)d_in[8];
    const int*   rows     = (const int*)d_in[9];
    const int*   cols     = (const int*)d_in[10];

    const size_t Hsz  = (size_t)SS * NN * OUTD;      // 20.48 M floats
    const size_t Zsz  = (size_t)NN * KK * OUTD;      // 23.04 M floats
    const size_t need = (Hsz + 3 * Zsz) * sizeof(float);
    if (ws_size < need) return;

    float* Hbuf  = (float*)d_ws;
    float* zbuf  = Hbuf + Hsz;
    float* zcbuf = zbuf + Zsz;
    float* zdbuf = zcbuf + Zsz;

    // 1) zero z (sector slots accumulated with atomics; slot 0 overwritten)
    {
        long n4 = (long)(Zsz / 4);
        zero_f4_kernel<<<dim3((unsigned)((n4 + 255) / 256)), 256, 0, stream>>>((float4*)zbuf, n4);
    }

    // 2) self projection: zbuf[:,0,:] = x @ W_self   (ldc = K*OUT)
    gemm_wmma_kernel<<<dim3(157, 1), 256, 0, stream>>>(
        x, IND, W_self, 0, zbuf, 0, KK * OUTD, nullptr, 1, NN, IND);

    // 3) sector projections: H[s] = (x @ W_sect[s]) * out_norm   (batched over s)
    gemm_wmma_kernel<<<dim3(157, SS), 256, 0, stream>>>(
        x, IND, W_sect, (long)IND * OUTD, Hbuf, (long)NN * OUTD, OUTD,
        out_norm, 1, NN, IND);

    // 4) edge scatter-add: zbuf[r, s+1, :] += H[s, c, :]
    agg_kernel<<<dim3((SS * EP) / 8), 256, 0, stream>>>(Hbuf, rows, cols, zbuf);

    // 5) zc = (z * in_norm) @ WC ; zd = (z * in_norm) @ WD
    //    (in_norm folded as output row-scale, node = row / 9)
    gemm_wmma_kernel<<<dim3(1407, 1), 256, 0, stream>>>(
        zbuf, OUTD, WC, 0, zcbuf, 0, OUTD, in_norm, KK, NN * KK, OUTD);
    gemm_wmma_kernel<<<dim3(1407, 1), 256, 0, stream>>>(
        zbuf, OUTD, WD, 0, zdbuf, 0, OUTD, in_norm, KK, NN * KK, OUTD);

    // 6) per-node interaction + gate + blend -> d_out
    interact_kernel<<<dim3(NN), 32, 0, stream>>>(zcbuf, zdbuf, gate_w, gate_b,
                                                 (float*)d_out);
}